// ScaleMultiHeadAttention_5712306504279
// MI455X (gfx1250) — compile-verified
//
#include <hip/hip_runtime.h>

typedef __attribute__((ext_vector_type(2))) float v2f;
typedef __attribute__((ext_vector_type(8))) float v8f;

#define SEQ   1024
#define MODEL 1024
#define DH    64
#define NBH   64      // BZ * N_HEADS
#define FDIM  2048    // MODEL * N_SCALE
#define ROWS  4096    // BZ * SEQ
#define Y_SIZE ((size_t)ROWS * FDIM)          // 8,388,608 floats
#define CT_STRIDE 128                          // per-seq stride in CT [bh][s][2][64]

__device__ __forceinline__ v8f wmma4(v2f a, v2f b, v8f c) {
  // D = A(16x4,f32) x B(4x16,f32) + C(16x16,f32)  -> v_wmma_f32_16x16x4_f32
  return __builtin_amdgcn_wmma_f32_16x16x4_f32(false, a, false, b, (short)0, c,
                                               false, false);
}

// ---------------------------------------------------------------------------
// NT GEMM: C[i][j] = sum_k A[i][k]*B[j][k] + bias[j] (+ addend[i][j])
// One wave computes a 16(M) x 64(N) tile. Depth-2 software pipeline: two
// 4-wide k stages live in distinct registers so up to 10 loads stream under
// the 8 WMMAs of the two in-flight stages.
// ---------------------------------------------------------------------------
__global__ __launch_bounds__(256, 1) void gemm_nt_kernel(
    const float* __restrict__ A, const float* __restrict__ B,
    const float* __restrict__ bias, const float* __restrict__ addend,
    float* __restrict__ C, int M, int N, int K)
{
  const int lane = threadIdx.x & 31;
  const int wid  = blockIdx.x * (blockDim.x >> 5) + (threadIdx.x >> 5);
  const int ntiles = N >> 6;
  const int mt = wid / ntiles;
  const int nt = wid % ntiles;
  if (mt * 16 >= M) return;

  const int lm = lane & 15;            // M (A) / N (B) index within tile
  const int lk = (lane >> 4) << 1;     // K pair selector: 0 or 2

  const float* Ap = A + (size_t)(mt * 16 + lm) * K + lk;
  const float* B0 = B + (size_t)(nt * 64 +  0 + lm) * K + lk;
  const float* B1 = B + (size_t)(nt * 64 + 16 + lm) * K + lk;
  const float* B2 = B + (size_t)(nt * 64 + 32 + lm) * K + lk;
  const float* B3 = B + (size_t)(nt * 64 + 48 + lm) * K + lk;

  v8f z = {};
  v8f acc0 = z, acc1 = z, acc2 = z, acc3 = z;

  // pipeline prologue: stage S0 (k=0) and S1 (k=4)
  v2f aS0  = *(const v2f*)(Ap);
  v2f b0S0 = *(const v2f*)(B0);
  v2f b1S0 = *(const v2f*)(B1);
  v2f b2S0 = *(const v2f*)(B2);
  v2f b3S0 = *(const v2f*)(B3);
  v2f aS1  = *(const v2f*)(Ap + 4);
  v2f b0S1 = *(const v2f*)(B0 + 4);
  v2f b1S1 = *(const v2f*)(B1 + 4);
  v2f b2S1 = *(const v2f*)(B2 + 4);
  v2f b3S1 = *(const v2f*)(B3 + 4);

  #pragma unroll 2
  for (int k0 = 0; k0 < K - 8; k0 += 8) {
    // consume S0, refill S0 with k0+8
    v2f a = aS0, b0 = b0S0, b1 = b1S0, b2 = b2S0, b3 = b3S0;
    aS0  = *(const v2f*)(Ap + k0 + 8);
    b0S0 = *(const v2f*)(B0 + k0 + 8);
    b1S0 = *(const v2f*)(B1 + k0 + 8);
    b2S0 = *(const v2f*)(B2 + k0 + 8);
    b3S0 = *(const v2f*)(B3 + k0 + 8);
    acc0 = wmma4(a, b0, acc0);
    acc1 = wmma4(a, b1, acc1);
    acc2 = wmma4(a, b2, acc2);
    acc3 = wmma4(a, b3, acc3);
    // consume S1, refill S1 with k0+12
    v2f c = aS1, d0 = b0S1, d1 = b1S1, d2 = b2S1, d3 = b3S1;
    aS1  = *(const v2f*)(Ap + k0 + 12);
    b0S1 = *(const v2f*)(B0 + k0 + 12);
    b1S1 = *(const v2f*)(B1 + k0 + 12);
    b2S1 = *(const v2f*)(B2 + k0 + 12);
    b3S1 = *(const v2f*)(B3 + k0 + 12);
    acc0 = wmma4(c, d0, acc0);
    acc1 = wmma4(c, d1, acc1);
    acc2 = wmma4(c, d2, acc2);
    acc3 = wmma4(c, d3, acc3);
  }
  // epilogue: last two stages
  acc0 = wmma4(aS0, b0S0, acc0);
  acc1 = wmma4(aS0, b1S0, acc1);
  acc2 = wmma4(aS0, b2S0, acc2);
  acc3 = wmma4(aS0, b3S0, acc3);
  acc0 = wmma4(aS1, b0S1, acc0);
  acc1 = wmma4(aS1, b1S1, acc1);
  acc2 = wmma4(aS1, b2S1, acc2);
  acc3 = wmma4(aS1, b3S1, acc3);

  const int mrow0 = mt * 16 + ((lane >> 4) << 3);
  const int ncol  = nt * 64 + lm;
  #pragma unroll
  for (int v = 0; v < 8; ++v) {
    const size_t base = (size_t)(mrow0 + v) * N + ncol;
    float o0 = acc0[v] + bias[ncol];
    float o1 = acc1[v] + bias[ncol + 16];
    float o2 = acc2[v] + bias[ncol + 32];
    float o3 = acc3[v] + bias[ncol + 48];
    if (addend) {
      o0 += addend[base];      o1 += addend[base + 16];
      o2 += addend[base + 32]; o3 += addend[base + 48];
    }
    C[base] = o0; C[base + 16] = o1; C[base + 32] = o2; C[base + 48] = o3;
  }
}

// ---------------------------------------------------------------------------
// Fused SDPA: one wave owns (bh, 16-query tile). Scores(16x1024) in LDS,
// softmax, attn1 DMA'd LDS->HBM via global_store_async_from_lds_b128
// (overlaps with the P@V WMMA phase), then ctx = P @ V via WMMA.
// scores = Q K^T * 8 (reference multiplies by sqrt(d_head)).
// ---------------------------------------------------------------------------
__global__ __launch_bounds__(32, 1) void sdpa_kernel(
    const float* __restrict__ Qbase, int q_ct_layout,
    const float* __restrict__ Kfull, const float* __restrict__ Vfull,
    float* __restrict__ attn_out,      // nullptr to skip
    float* __restrict__ ctx_base)      // CT (+64 for pass 2), stride 128
{
  __shared__ float P[16 * SEQ];        // 64 KB

  const int qt   = blockIdx.x;         // 0..63 query tile
  const int bh   = blockIdx.y;         // 0..63 batch-head
  const int b    = bh >> 4;
  const int h    = bh & 15;
  const int lane = threadIdx.x;
  const int lm   = lane & 15;
  const int lk   = (lane >> 4) << 1;

  const float* Q = q_ct_layout
      ? (Qbase + (size_t)bh * SEQ * CT_STRIDE)
      : (Qbase + (size_t)b * SEQ * MODEL + h * DH);
  const int qstride = q_ct_layout ? CT_STRIDE : MODEL;
  const float* Kh = Kfull + (size_t)b * SEQ * MODEL + h * DH;
  const float* Vh = Vfull + (size_t)b * SEQ * MODEL + h * DH;

  // Preload the 16 A fragments covering d = 0..63 of this wave's Q rows.
  v2f qa[16];
  {
    const float* qrow = Q + (size_t)(qt * 16 + lm) * qstride + lk;
    #pragma unroll
    for (int kk = 0; kk < 16; ++kk)
      qa[kk] = *(const v2f*)(qrow + kk * 4);
  }

  v8f z8 = {};

  // ---- scores: 64 key tiles of 16, double-buffered K fragments ----
  v2f kbuf[16];
  {
    const float* kr = Kh + (size_t)lm * MODEL + lk;
    #pragma unroll
    for (int kk = 0; kk < 16; ++kk) kbuf[kk] = *(const v2f*)(kr + kk * 4);
  }
  for (int kt = 0; kt < 64; ++kt) {
    v2f kcur[16];
    #pragma unroll
    for (int i = 0; i < 16; ++i) kcur[i] = kbuf[i];
    if (kt < 63) {
      const float* kr = Kh + (size_t)((kt + 1) * 16 + lm) * MODEL + lk;
      #pragma unroll
      for (int kk = 0; kk < 16; ++kk) kbuf[kk] = *(const v2f*)(kr + kk * 4);
    }
    v8f accA = z8, accB = z8;            // two chains break D->C serialization
    #pragma unroll
    for (int kk = 0; kk < 16; kk += 2) {
      accA = wmma4(qa[kk],     kcur[kk],     accA);
      accB = wmma4(qa[kk + 1], kcur[kk + 1], accB);
    }
    #pragma unroll
    for (int v = 0; v < 8; ++v) {
      int m = ((lane >> 4) << 3) + v;
      P[m * SEQ + kt * 16 + lm] = (accA[v] + accB[v]) * 8.0f;  // scale=sqrt(64)
    }
  }
  __syncthreads();

  // ---- row softmax over 1024 keys (float4-vectorized, wave32 shuffles) ----
  for (int m = 0; m < 16; ++m) {
    float4* row4 = (float4*)&P[m * SEQ];
    float mx = -3.402823466e38f;
    #pragma unroll 2
    for (int c = lane; c < SEQ / 4; c += 32) {
      float4 v = row4[c];
      mx = fmaxf(mx, fmaxf(fmaxf(v.x, v.y), fmaxf(v.z, v.w)));
    }
    #pragma unroll
    for (int off = 16; off; off >>= 1) mx = fmaxf(mx, __shfl_xor(mx, off, 32));
    float sum = 0.0f;
    #pragma unroll 2
    for (int c = lane; c < SEQ / 4; c += 32) {
      float4 v = row4[c];
      v.x = __expf(v.x - mx); v.y = __expf(v.y - mx);
      v.z = __expf(v.z - mx); v.w = __expf(v.w - mx);
      row4[c] = v;
      sum += (v.x + v.y) + (v.z + v.w);
    }
    #pragma unroll
    for (int off = 16; off; off >>= 1) sum += __shfl_xor(sum, off, 32);
    float inv = 1.0f / sum;
    #pragma unroll 2
    for (int c = lane; c < SEQ / 4; c += 32) {
      float4 v = row4[c];
      v.x *= inv; v.y *= inv; v.z *= inv; v.w *= inv;
      row4[c] = v;
    }
  }
  __syncthreads();

  // ---- attn output (pass 1): async DMA LDS -> HBM, 512B per issue ----
  if (attn_out) {
    asm volatile("s_wait_dscnt 0x0" ::: "memory");   // LDS writes visible
    const unsigned lbase = (unsigned)(size_t)(&P[0]); // low 32 bits = LDS offset
    const unsigned long long gbase = (unsigned long long)(size_t)
        (attn_out + ((size_t)bh * SEQ + qt * 16) * SEQ);
    for (int m = 0; m < 16; ++m) {
      const unsigned lrow = lbase + (unsigned)(m * SEQ * 4) + lane * 16;
      const unsigned long long grow =
          gbase + (unsigned long long)(m * SEQ * 4) + lane * 16;
      #pragma unroll
      for (int ch = 0; ch < 8; ++ch) {
        unsigned laddr = lrow + ch * 512;
        unsigned long long gaddr = grow + (unsigned long long)(ch * 512);
        asm volatile("global_store_async_from_lds_b128 %0, %1, off"
                     :: "v"(gaddr), "v"(laddr) : "memory");
      }
    }
    // no wait needed here: the P@V phase only READS LDS; s_endpgm waits all.
  }

  // ---- ctx = P(16x1024) @ V(1024x64), double-buffered V fragments ----
  v8f cacc0 = z8, cacc1 = z8, cacc2 = z8, cacc3 = z8;
  auto loadV = [&](int kt, v2f* vb) {
    #pragma unroll
    for (int kk = 0; kk < 4; ++kk) {
      const float* v0 = Vh + (size_t)(kt * 16 + kk * 4 + lk) * MODEL;
      #pragma unroll
      for (int n0 = 0; n0 < 4; ++n0) {
        v2f t;
        t.x = v0[n0 * 16 + lm];
        t.y = v0[MODEL + n0 * 16 + lm];
        vb[kk * 4 + n0] = t;
      }
    }
  };
  v2f vbuf[16];
  loadV(0, vbuf);
  for (int kt = 0; kt < 64; ++kt) {
    v2f vcur[16];
    #pragma unroll
    for (int i = 0; i < 16; ++i) vcur[i] = vbuf[i];
    if (kt < 63) loadV(kt + 1, vbuf);
    #pragma unroll
    for (int kk = 0; kk < 4; ++kk) {
      v2f a = *(const v2f*)&P[lm * SEQ + kt * 16 + kk * 4 + lk];
      cacc0 = wmma4(a, vcur[kk * 4 + 0], cacc0);
      cacc1 = wmma4(a, vcur[kk * 4 + 1], cacc1);
      cacc2 = wmma4(a, vcur[kk * 4 + 2], cacc2);
      cacc3 = wmma4(a, vcur[kk * 4 + 3], cacc3);
    }
  }

  float* cp = ctx_base + (size_t)bh * SEQ * CT_STRIDE
                       + (size_t)(qt * 16) * CT_STRIDE;
  #pragma unroll
  for (int v = 0; v < 8; ++v) {
    int m = ((lane >> 4) << 3) + v;
    cp[(size_t)m * CT_STRIDE +  0 + lm] = cacc0[v];
    cp[(size_t)m * CT_STRIDE + 16 + lm] = cacc1[v];
    cp[(size_t)m * CT_STRIDE + 32 + lm] = cacc2[v];
    cp[(size_t)m * CT_STRIDE + 48 + lm] = cacc3[v];
  }
}

// ---------------------------------------------------------------------------
// Row LayerNorm over 2048 features (float4-vectorized).
// ---------------------------------------------------------------------------
__global__ __launch_bounds__(256) void ln_kernel(
    const float* __restrict__ X, const float* __restrict__ gamma,
    const float* __restrict__ beta, float* __restrict__ Y)
{
  __shared__ float s1[8], s2[8];
  const int row = blockIdx.x;
  const float4* x4 = (const float4*)(X + (size_t)row * FDIM);
  float s = 0.0f, sq = 0.0f;
  #pragma unroll
  for (int c = threadIdx.x; c < FDIM / 4; c += 256) {
    float4 v = x4[c];
    s  += (v.x + v.y) + (v.z + v.w);
    sq += (v.x * v.x + v.y * v.y) + (v.z * v.z + v.w * v.w);
  }
  #pragma unroll
  for (int off = 16; off; off >>= 1) {
    s  += __shfl_xor(s,  off, 32);
    sq += __shfl_xor(sq, off, 32);
  }
  const int wid = threadIdx.x >> 5;
  if ((threadIdx.x & 31) == 0) { s1[wid] = s; s2[wid] = sq; }
  __syncthreads();
  float ts = 0.0f, tsq = 0.0f;
  #pragma unroll
  for (int i = 0; i < 8; ++i) { ts += s1[i]; tsq += s2[i]; }
  const float mu  = ts * (1.0f / FDIM);
  const float var = tsq * (1.0f / FDIM) - mu * mu;
  const float inv = rsqrtf(var + 1e-5f);
  const float4* g4 = (const float4*)gamma;
  const float4* b4 = (const float4*)beta;
  float4* y4 = (float4*)(Y + (size_t)row * FDIM);
  #pragma unroll
  for (int c = threadIdx.x; c < FDIM / 4; c += 256) {
    float4 v = x4[c], g = g4[c], bb = b4[c], o;
    o.x = (v.x - mu) * inv * g.x + bb.x;
    o.y = (v.y - mu) * inv * g.y + bb.y;
    o.z = (v.z - mu) * inv * g.z + bb.z;
    o.w = (v.w - mu) * inv * g.w + bb.w;
    y4[c] = o;
  }
}

// ---------------------------------------------------------------------------
extern "C" void kernel_launch(void* const* d_in, const int* in_sizes, int n_in,
                              void* d_out, int out_size, void* d_ws, size_t ws_size,
                              hipStream_t stream)
{
  (void)in_sizes; (void)n_in; (void)out_size; (void)ws_size;
  const float* k_in  = (const float*)d_in[0];
  const float* v_in  = (const float*)d_in[1];
  const float* q_in  = (const float*)d_in[2];
  const float* r_in  = (const float*)d_in[3];
  const float* Wr    = (const float*)d_in[4];
  const float* br    = (const float*)d_in[5];
  const float* Wf    = (const float*)d_in[6];
  const float* bf    = (const float*)d_in[7];
  const float* gamma = (const float*)d_in[8];
  const float* beta  = (const float*)d_in[9];

  float* y_out    = (float*)d_out;
  float* attn_out = y_out + Y_SIZE;                 // attn1 [64,1024,1024]

  float* rp = (float*)d_ws;                          // [4096][2048]  32 MB
  float* CT = rp + (size_t)ROWS * FDIM;              // [64][1024][2][64] 32 MB

  // 1) rp = r @ Wr^T + br      (M=4096, N=2048, K=1024)
  {
    int waves  = (ROWS / 16) * (FDIM / 64);          // 8192
    int blocks = waves / 8;
    gemm_nt_kernel<<<blocks, 256, 0, stream>>>(r_in, Wr, br, nullptr, rp,
                                               ROWS, FDIM, MODEL);
  }

  // 2) SDPA pass 1: attn1 to d_out, ctx1 -> CT (p=0)
  dim3 sg(SEQ / 16, NBH);
  sdpa_kernel<<<sg, 32, 0, stream>>>(q_in, 0, k_in, v_in, attn_out, CT);

  // 3) SDPA pass 2: queries = ctx1 (CT layout), ctx2 -> CT + 64 (p=1)
  sdpa_kernel<<<sg, 32, 0, stream>>>(CT, 1, k_in, v_in, nullptr, CT + DH);

  // 4) x = rp + CT @ Wf^T + bf  (in-place into rp; M=4096, N=2048, K=2048)
  {
    int waves  = (ROWS / 16) * (FDIM / 64);
    int blocks = waves / 8;
    gemm_nt_kernel<<<blocks, 256, 0, stream>>>(CT, Wf, bf, rp, rp,
                                               ROWS, FDIM, FDIM);
  }

  // 5) LayerNorm rows -> y
  ln_kernel<<<ROWS, 256, 0, stream>>>(rp, gamma, beta, y_out);
}